// Net_41944650613290
// MI455X (gfx1250) — compile-verified
//
#include <hip/hip_runtime.h>
#include <hip/hip_bf16.h>
#include <math.h>

typedef __attribute__((ext_vector_type(2))) float v2f;
typedef __attribute__((ext_vector_type(8))) float v8f;

#define BATCH  2
#define NN     128
#define NE     160
#define NHEADS 8
#define DHEAD  64
#define INNER  512
#define DIMM   128
#define ROWS   (BATCH*NN)   // 256

// ---------------------------------------------------------------- utilities

__global__ void zero_kernel(float* p, int n) {
  int i = blockIdx.x * blockDim.x + threadIdx.x;
  int stride = gridDim.x * blockDim.x;
  for (; i < n; i += stride) p[i] = 0.0f;
}

__global__ void build_nodes_kernel(const int* __restrict__ idx,
                                   const float* __restrict__ emb,
                                   const float* __restrict__ noise,
                                   float* __restrict__ nodes) {
  int row = blockIdx.x;     // b*128 + n
  int c   = threadIdx.x;    // 0..127
  int a   = idx[row];
  float v = (c < 127) ? emb[a * 127 + c] : noise[0];
  nodes[(size_t)row * DIMM + c] = v;
}

// sign=+1: edges[b,i,j]=diff ; sign=-1: edges[b,j,i]=-diff (two passes match ref order)
__global__ void edges_set_kernel(const int* __restrict__ bonds,
                                 const float* __restrict__ coords,
                                 float* __restrict__ edges, int sign) {
  int b = blockIdx.x;
  int e = threadIdx.x;
  if (e >= NE) return;
  int i = bonds[e * 2 + 0], j = bonds[e * 2 + 1];
  const float* ci = coords + ((size_t)b * NN + i) * 3;
  const float* cj = coords + ((size_t)b * NN + j) * 3;
  float d0 = ci[0] - cj[0], d1 = ci[1] - cj[1], d2 = ci[2] - cj[2];
  int ii = (sign > 0) ? i : j;
  int jj = (sign > 0) ? j : i;
  float s = (float)sign;
  float* dst = edges + (((size_t)b * NN + ii) * NN + jj) * 3;
  dst[0] = s * d0; dst[1] = s * d1; dst[2] = s * d2;
}

__global__ void ln_kernel(const float* __restrict__ x,
                          const float* __restrict__ g,
                          const float* __restrict__ bp,
                          float* __restrict__ y) {
  int row = blockIdx.x;
  int c   = threadIdx.x;         // 0..127
  __shared__ float red[DIMM];
  float v = x[(size_t)row * DIMM + c];
  red[c] = v; __syncthreads();
  for (int off = 64; off > 0; off >>= 1) { if (c < off) red[c] += red[c + off]; __syncthreads(); }
  float m = red[0] * (1.0f / DIMM); __syncthreads();
  float d = v - m;
  red[c] = d * d; __syncthreads();
  for (int off = 64; off > 0; off >>= 1) { if (c < off) red[c] += red[c + off]; __syncthreads(); }
  float var = red[0] * (1.0f / DIMM);
  y[(size_t)row * DIMM + c] = d * rsqrtf(var + 1e-5f) * g[c] + bp[c];
}

// ------------------------------------------------ generic f32 WMMA GEMM
// C[M,N] = act(A[M,K] @ W[K,N] + bias[N]); one wave per 16x16 tile, K step 4.
template <int ACT>
__global__ void wmma_gemm_kernel(const float* __restrict__ A,
                                 const float* __restrict__ W,
                                 const float* __restrict__ bias,
                                 float* __restrict__ C,
                                 int M, int N, int K) {
  int wave = threadIdx.x >> 5, lane = threadIdx.x & 31;
  int tile = blockIdx.x * 4 + wave;
  int tilesN = N >> 4;
  if (tile >= (M >> 4) * tilesN) return;       // wave-uniform
  int tm = tile / tilesN, tn = tile - tm * tilesN;
  int half = lane >> 4, lo = lane & 15;
  const float* arow = A + (size_t)(tm * 16 + lo) * K;
  const float* bcol = W + tn * 16 + lo;
  v8f acc = {};
  for (int k0 = 0; k0 < K; k0 += 4) {
    int ka = k0 + half * 2;
    v2f a, bb;
    a[0]  = arow[ka];
    a[1]  = arow[ka + 1];
    bb[0] = bcol[(size_t)ka * N];
    bb[1] = bcol[(size_t)(ka + 1) * N];
    acc = __builtin_amdgcn_wmma_f32_16x16x4_f32(false, a, false, bb, (short)0, acc, false, false);
  }
  int n = tn * 16 + lo;
  float bv = bias[n];
  for (int r = 0; r < 8; ++r) {
    int m = tm * 16 + half * 8 + r;
    float v = acc[r] + bv;
    if (ACT == 1) v = 0.5f * v * (1.0f + erff(v * 0.70710678118654752f)); // exact GELU
    C[(size_t)m * N + n] = v;
  }
}

// -------------------------------------------- simqk[b,h,i,j] = q_i . k_j
__global__ void qk_kernel(const float* __restrict__ q,
                          const float* __restrict__ kv,   // k = first 512 of 1024
                          float* __restrict__ simqk) {
  int wave = threadIdx.x >> 5, lane = threadIdx.x & 31;
  int t  = blockIdx.x * 4 + wave;  // 0..1023 : ((b*8+h)*64 + ti*8 + tj)
  int bh = t >> 6, tt = t & 63;
  int b = bh >> 3, h = bh & 7;
  int ti = tt >> 3, tj = tt & 7;
  int half = lane >> 4, lo = lane & 15;
  const float* qrow = q  + (size_t)(b * NN + ti * 16 + lo) * INNER + h * DHEAD;
  const float* krow = kv + (size_t)(b * NN + tj * 16 + lo) * (2 * INNER) + h * DHEAD;
  v8f acc = {};
  for (int d0 = 0; d0 < DHEAD; d0 += 4) {
    int dd = d0 + half * 2;
    v2f a, bb;
    a[0] = qrow[dd];  a[1] = qrow[dd + 1];
    bb[0] = krow[dd]; bb[1] = krow[dd + 1];
    acc = __builtin_amdgcn_wmma_f32_16x16x4_f32(false, a, false, bb, (short)0, acc, false, false);
  }
  float* out = simqk + (size_t)bh * NN * NN;
  for (int r = 0; r < 8; ++r)
    out[(size_t)(ti * 16 + half * 8 + r) * NN + tj * 16 + lo] = acc[r];
}

// ------------------------- qWe[b,i,h,c] = q.We[c], qbe[b,i,h] = q.be (per head)
__global__ void qaux_kernel(const float* __restrict__ q,
                            const float* __restrict__ We,
                            const float* __restrict__ be,
                            float* __restrict__ qWe, float* __restrict__ qbe) {
  int t = blockIdx.x * blockDim.x + threadIdx.x;   // (b*128+i)*8+h
  if (t >= ROWS * NHEADS) return;
  int h = t & 7, bi = t >> 3;
  const float* qp = q + (size_t)bi * INNER + h * DHEAD;
  const float* w0 = We + h * DHEAD;
  float s0 = 0, s1 = 0, s2 = 0, sb = 0;
  for (int d = 0; d < DHEAD; ++d) {
    float qv = qp[d];
    s0 += qv * w0[d];
    s1 += qv * w0[INNER + d];
    s2 += qv * w0[2 * INNER + d];
    sb += qv * be[h * DHEAD + d];
  }
  qWe[t * 3 + 0] = s0; qWe[t * 3 + 1] = s1; qWe[t * 3 + 2] = s2;
  qbe[t] = sb;
}

// -------- softmax over j with edge correction; also aE[b,i,h,c] = sum_j attn*edge_c
__global__ void softmax_edge_kernel(const float* __restrict__ simqk,
                                    const float* __restrict__ edges,
                                    const float* __restrict__ qWe,
                                    const float* __restrict__ qbe,
                                    float* __restrict__ attn,
                                    float* __restrict__ aE) {
  int idx = blockIdx.x;          // (b*8+h)*128 + i
  int i = idx & 127, bh = idx >> 7;
  int b = bh >> 3, h = bh & 7;
  int j = threadIdx.x;
  __shared__ float red[NN];
  const float* ed = edges + (((size_t)b * NN + i) * NN + j) * 3;
  const float* qw = qWe + (((size_t)b * NN + i) * NHEADS + h) * 3;
  float s = simqk[((size_t)bh * NN + i) * NN + j]
          + ed[0] * qw[0] + ed[1] * qw[1] + ed[2] * qw[2]
          + qbe[((size_t)b * NN + i) * NHEADS + h];
  s *= 0.125f;                   // DIM_HEAD^-0.5
  red[j] = s; __syncthreads();
  for (int off = 64; off > 0; off >>= 1) { if (j < off) red[j] = fmaxf(red[j], red[j + off]); __syncthreads(); }
  float mx = red[0]; __syncthreads();
  float e = expf(s - mx);
  red[j] = e; __syncthreads();
  for (int off = 64; off > 0; off >>= 1) { if (j < off) red[j] += red[j + off]; __syncthreads(); }
  float a = e / red[0]; __syncthreads();
  attn[((size_t)bh * NN + i) * NN + j] = a;
  for (int c = 0; c < 3; ++c) {
    red[j] = a * ed[c]; __syncthreads();
    for (int off = 64; off > 0; off >>= 1) { if (j < off) red[j] += red[j + off]; __syncthreads(); }
    if (j == 0) aE[(((size_t)b * NN + i) * NHEADS + h) * 3 + c] = red[0];
    __syncthreads();
  }
}

// ------------------------------------ hout[b,i,h*64+d] = sum_j attn[b,h,i,j]*v[b,j,h,d]
__global__ void attnv_kernel(const float* __restrict__ attn,
                             const float* __restrict__ kv,  // v = second 512 of 1024
                             float* __restrict__ hout) {
  int wave = threadIdx.x >> 5, lane = threadIdx.x & 31;
  int t  = blockIdx.x * 4 + wave;  // 0..511 : ((b*8+h)*32 + ti*4 + tn)
  int bh = t >> 5, tt = t & 31;
  int b = bh >> 3, h = bh & 7;
  int ti = tt >> 2, tn = tt & 3;
  int half = lane >> 4, lo = lane & 15;
  const float* arow  = attn + ((size_t)bh * NN + ti * 16 + lo) * NN;
  const float* vbase = kv + INNER + h * DHEAD + tn * 16 + lo;
  v8f acc = {};
  for (int j0 = 0; j0 < NN; j0 += 4) {
    int jj = j0 + half * 2;
    v2f a, bb;
    a[0]  = arow[jj];
    a[1]  = arow[jj + 1];
    bb[0] = vbase[(size_t)(b * NN + jj) * (2 * INNER)];
    bb[1] = vbase[(size_t)(b * NN + jj + 1) * (2 * INNER)];
    acc = __builtin_amdgcn_wmma_f32_16x16x4_f32(false, a, false, bb, (short)0, acc, false, false);
  }
  for (int r = 0; r < 8; ++r)
    hout[(size_t)(b * NN + ti * 16 + half * 8 + r) * INNER + h * DHEAD + tn * 16 + lo] = acc[r];
}

// hout += aE . We + be   (rank-3 edge-value correction; softmax rows sum to 1)
__global__ void edge_out_kernel(const float* __restrict__ aE,
                                const float* __restrict__ We,
                                const float* __restrict__ be,
                                float* __restrict__ hout) {
  int row = blockIdx.x;          // b*128 + i
  int hd  = threadIdx.x;         // 0..511
  int h   = hd >> 6;
  const float* a = aE + ((size_t)row * NHEADS + h) * 3;
  hout[(size_t)row * INNER + hd] +=
      a[0] * We[hd] + a[1] * We[INNER + hd] + a[2] * We[2 * INNER + hd] + be[hd];
}

__global__ void gated_residual_kernel(const float* __restrict__ x,
                                      float* __restrict__ nodes,
                                      const float* __restrict__ Wg) {
  int row = blockIdx.x;
  int c   = threadIdx.x;
  __shared__ float red[DIMM];
  float xv = x[(size_t)row * DIMM + c];
  float rv = nodes[(size_t)row * DIMM + c];
  red[c] = xv * Wg[c] + rv * Wg[DIMM + c] + (xv - rv) * Wg[2 * DIMM + c];
  __syncthreads();
  for (int off = 64; off > 0; off >>= 1) { if (c < off) red[c] += red[c + off]; __syncthreads(); }
  float gate = 1.0f / (1.0f + expf(-red[0]));
  nodes[(size_t)row * DIMM + c] = xv * gate + rv * (1.0f - gate);
}

__global__ void final_kernel(const float* __restrict__ nodes,
                             const float* __restrict__ Wlin,
                             const float* __restrict__ blin,
                             float* __restrict__ out) {
  __shared__ float red[ROWS];
  int t = threadIdx.x;           // 0..255 : one row each
  const float* row = nodes + (size_t)t * DIMM;
  float acc = 0.0f;
  for (int c = 0; c < DIMM; ++c) acc += row[c] * Wlin[c];
  red[t] = acc; __syncthreads();
  for (int off = 128; off > 0; off >>= 1) { if (t < off) red[t] += red[t + off]; __syncthreads(); }
  if (t == 0) out[0] = red[0] + (float)ROWS * blin[0];
}

// ---------------------------------------------------------------- host

extern "C" void kernel_launch(void* const* d_in, const int* in_sizes, int n_in,
                              void* d_out, int out_size, void* d_ws, size_t ws_size,
                              hipStream_t stream) {
  const int*   indices  = (const int*)  d_in[0];
  const float* coords   = (const float*)d_in[1];
  const int*   bonds    = (const int*)  d_in[2];
  const float* noise    = (const float*)d_in[3];
  const float* atom_emb = (const float*)d_in[4];
  const float* ln1_g = (const float*)d_in[5];
  const float* ln1_b = (const float*)d_in[6];
  const float* Wq    = (const float*)d_in[7];
  const float* bq    = (const float*)d_in[8];
  const float* Wkv   = (const float*)d_in[9];
  const float* bkv   = (const float*)d_in[10];
  const float* We    = (const float*)d_in[11];
  const float* be    = (const float*)d_in[12];
  const float* Wo    = (const float*)d_in[13];
  const float* bo    = (const float*)d_in[14];
  const float* Wg1   = (const float*)d_in[15];
  const float* ln2_g = (const float*)d_in[16];
  const float* ln2_b = (const float*)d_in[17];
  const float* W1    = (const float*)d_in[18];
  const float* b1    = (const float*)d_in[19];
  const float* W2    = (const float*)d_in[20];
  const float* b2    = (const float*)d_in[21];
  const float* Wg2   = (const float*)d_in[22];
  const float* Wlin  = (const float*)d_in[23];
  const float* blin  = (const float*)d_in[24];
  (void)in_sizes; (void)n_in; (void)out_size; (void)ws_size;

  float* ws = (float*)d_ws;
  size_t o = 0;
  float* nodes = ws + o; o += (size_t)ROWS * DIMM;
  float* edges = ws + o; o += (size_t)BATCH * NN * NN * 3;
  float* xn    = ws + o; o += (size_t)ROWS * DIMM;
  float* qb    = ws + o; o += (size_t)ROWS * INNER;
  float* kvb   = ws + o; o += (size_t)ROWS * 2 * INNER;
  float* simqk = ws + o; o += (size_t)BATCH * NHEADS * NN * NN;
  float* attnb = ws + o; o += (size_t)BATCH * NHEADS * NN * NN;
  float* qWeb  = ws + o; o += (size_t)ROWS * NHEADS * 3;
  float* qbeb  = ws + o; o += (size_t)ROWS * NHEADS;
  float* aEb   = ws + o; o += (size_t)ROWS * NHEADS * 3;
  float* hout  = ws + o; o += (size_t)ROWS * INNER;
  float* proj  = ws + o; o += (size_t)ROWS * DIMM;
  float* ff1   = ws + o; o += (size_t)ROWS * INNER;

  build_nodes_kernel<<<ROWS, DIMM, 0, stream>>>(indices, atom_emb, noise, nodes);
  zero_kernel<<<96, 256, 0, stream>>>(edges, BATCH * NN * NN * 3);
  edges_set_kernel<<<BATCH, NE, 0, stream>>>(bonds, coords, edges, +1);
  edges_set_kernel<<<BATCH, NE, 0, stream>>>(bonds, coords, edges, -1);

  for (int l = 0; l < 6; ++l) {
    const float* g1    = ln1_g + l * DIMM;
    const float* bb1   = ln1_b + l * DIMM;
    const float* Wq_l  = Wq  + (size_t)l * DIMM * INNER;
    const float* bq_l  = bq  + l * INNER;
    const float* Wkv_l = Wkv + (size_t)l * DIMM * 2 * INNER;
    const float* bkv_l = bkv + l * 2 * INNER;
    const float* We_l  = We  + (size_t)l * 3 * INNER;
    const float* be_l  = be  + l * INNER;
    const float* Wo_l  = Wo  + (size_t)l * INNER * DIMM;
    const float* bo_l  = bo  + l * DIMM;
    const float* Wg1_l = Wg1 + l * 3 * DIMM;
    const float* g2    = ln2_g + l * DIMM;
    const float* bb2   = ln2_b + l * DIMM;
    const float* W1_l  = W1  + (size_t)l * DIMM * 4 * DIMM;
    const float* b1_l  = b1  + l * 4 * DIMM;
    const float* W2_l  = W2  + (size_t)l * 4 * DIMM * DIMM;
    const float* b2_l  = b2  + l * DIMM;
    const float* Wg2_l = Wg2 + l * 3 * DIMM;

    ln_kernel<<<ROWS, DIMM, 0, stream>>>(nodes, g1, bb1, xn);
    // q = xn @ Wq + bq : (256 x 512, K=128) -> 512 tiles / 4 waves
    wmma_gemm_kernel<0><<<128, 128, 0, stream>>>(xn, Wq_l, bq_l, qb, ROWS, INNER, DIMM);
    // kv = xn @ Wkv + bkv : (256 x 1024, K=128) -> 1024 tiles
    wmma_gemm_kernel<0><<<256, 128, 0, stream>>>(xn, Wkv_l, bkv_l, kvb, ROWS, 2 * INNER, DIMM);
    qk_kernel<<<256, 128, 0, stream>>>(qb, kvb, simqk);
    qaux_kernel<<<16, 128, 0, stream>>>(qb, We_l, be_l, qWeb, qbeb);
    softmax_edge_kernel<<<BATCH * NHEADS * NN, NN, 0, stream>>>(simqk, edges, qWeb, qbeb, attnb, aEb);
    attnv_kernel<<<128, 128, 0, stream>>>(attnb, kvb, hout);
    edge_out_kernel<<<ROWS, INNER, 0, stream>>>(aEb, We_l, be_l, hout);
    // proj = hout @ Wo + bo : (256 x 128, K=512) -> 128 tiles
    wmma_gemm_kernel<0><<<32, 128, 0, stream>>>(hout, Wo_l, bo_l, proj, ROWS, DIMM, INNER);
    gated_residual_kernel<<<ROWS, DIMM, 0, stream>>>(proj, nodes, Wg1_l);
    ln_kernel<<<ROWS, DIMM, 0, stream>>>(nodes, g2, bb2, xn);
    // ff1 = gelu(xn @ W1 + b1) : (256 x 512, K=128)
    wmma_gemm_kernel<1><<<128, 128, 0, stream>>>(xn, W1_l, b1_l, ff1, ROWS, 4 * DIMM, DIMM);
    // ff = ff1 @ W2 + b2 : (256 x 128, K=512)
    wmma_gemm_kernel<0><<<32, 128, 0, stream>>>(ff1, W2_l, b2_l, proj, ROWS, DIMM, 4 * DIMM);
    gated_residual_kernel<<<ROWS, DIMM, 0, stream>>>(proj, nodes, Wg2_l);
  }

  final_kernel<<<1, ROWS, 0, stream>>>(nodes, Wlin, blin, (float*)d_out);
}